// ConsistencyLoss_28776280883544
// MI455X (gfx1250) — compile-verified
//
#include <hip/hip_runtime.h>
#include <hip/hip_bf16.h>

#define IMG_H 1024
#define IMG_W 1024
#define TILE_X 128
#define TILE_Y 16
#define HALO 2
#define LDS_W (TILE_X + 2*HALO)   /* 132 */
#define LDS_H (TILE_Y + 2*HALO)   /* 20  */
#define NTHREADS 256

typedef float v2f __attribute__((ext_vector_type(2)));
typedef float v8f __attribute__((ext_vector_type(8)));

__global__ void closs_init(double* acc) {
    if (threadIdx.x == 0) acc[0] = 0.0;
}

__global__ __launch_bounds__(NTHREADS)
void closs_main(const float* __restrict__ ytrue,
                const float* __restrict__ ypred,
                double* __restrict__ acc) {
    __shared__ float tile[LDS_H][LDS_W];
    __shared__ float red[NTHREADS / 32];

    const int tid = threadIdx.x;
    const int gx0 = blockIdx.x * TILE_X;
    const int gy0 = blockIdx.y * TILE_Y;
    const size_t img = (size_t)blockIdx.z * ((size_t)IMG_H * IMG_W);

    // ---- Stage y_pred tile (+halo) into LDS via gfx1250 async loads ----
    // In-bounds slots: global_load_async_to_lds_b32 (ASYNCcnt path).
    // Out-of-bounds halo slots: plain LDS store of 0 (zero padding).
    // The two paths never touch the same LDS address -> no ordering hazard.
    for (int s = tid; s < LDS_H * LDS_W; s += NTHREADS) {
        const int ly = s / LDS_W;
        const int lx = s - ly * LDS_W;
        const int gy = gy0 + ly - HALO;
        const int gx = gx0 + lx - HALO;
        if ((unsigned)gy < (unsigned)IMG_H && (unsigned)gx < (unsigned)IMG_W) {
            const float* gsrc = ypred + img + (size_t)gy * IMG_W + gx;
            // Low 32 bits of a generic pointer to LDS == LDS byte address
            // (flat LDS aperture).
            unsigned ldsaddr = (unsigned)(unsigned long long)&tile[ly][lx];
            asm volatile("global_load_async_to_lds_b32 %0, %1, off"
                         :: "v"(ldsaddr), "v"(gsrc) : "memory");
        } else {
            tile[ly][lx] = 0.0f;
        }
    }
    asm volatile("s_wait_asynccnt 0" ::: "memory");
    __syncthreads();

    // ---- Compute 8 elements per thread ----
    float part = 0.0f;
#pragma unroll
    for (int i = 0; i < (TILE_X * TILE_Y) / NTHREADS; ++i) {
        const int e  = tid + i * NTHREADS;
        const int ey = e >> 7;            // / TILE_X
        const int ex = e & (TILE_X - 1);  // % TILE_X
        const float c = tile[ey + HALO][ex + HALO];

        float w = fabsf(c - tile[ey][ex]);                     // (-2,-2)
        w = fmaxf(w, fabsf(c - tile[ey][ex + 2]));             // (-2, 0)
        w = fmaxf(w, fabsf(c - tile[ey][ex + 4]));             // (-2,+2)
        w = fmaxf(w, fabsf(c - tile[ey + 2][ex]));             // ( 0,-2)
        w = fmaxf(w, fabsf(c - tile[ey + 2][ex + 4]));         // ( 0,+2)
        w = fmaxf(w, fabsf(c - tile[ey + 4][ex]));             // (+2,-2)
        w = fmaxf(w, fabsf(c - tile[ey + 4][ex + 2]));         // (+2, 0)
        w = fmaxf(w, fabsf(c - tile[ey + 4][ex + 4]));         // (+2,+2)

        const float yt  = ytrue[img + (size_t)(gy0 + ey) * IMG_W + (gx0 + ex)];
        const float lp  = fmaxf(logf(c), -100.0f);
        const float l1p = fmaxf(log1pf(-c), -100.0f);
        const float a   = -(yt * lp + (1.0f - yt) * l1p);
        const float th  = (c >= 0.5f) ? c : 0.0f;
        part += w * th + a;
    }

    // ---- Per-wave reduction with V_WMMA_F32_16X16X4_F32 ----
    // A = all-ones 16x4  =>  D[m][n] = sum_k B[k][n].  B holds the 32 lane
    // partials in VGPR0 (VGPR1 = 0), so the 16 column sums jointly cover all
    // 32 partials.  Every lane's d[0] is one column sum; a 16-lane butterfly
    // sum yields the wave total in every lane.
    v2f av; av.x = 1.0f; av.y = 1.0f;
    v2f bv; bv.x = part; bv.y = 0.0f;
    v8f cz = {};
    v8f d = __builtin_amdgcn_wmma_f32_16x16x4_f32(
        /*neg_a=*/false, av, /*neg_b=*/false, bv,
        /*c_mod=*/(short)0, cz, /*reuse_a=*/false, /*reuse_b=*/false);

    float t = d[0];
    t += __shfl_xor(t, 1, 32);
    t += __shfl_xor(t, 2, 32);
    t += __shfl_xor(t, 4, 32);
    t += __shfl_xor(t, 8, 32);

    if ((tid & 31) == 0) red[tid >> 5] = t;
    __syncthreads();
    if (tid == 0) {
        float bsum = 0.0f;
#pragma unroll
        for (int i = 0; i < NTHREADS / 32; ++i) bsum += red[i];
        atomicAdd(acc, (double)bsum);   // global_atomic_add_f64
    }
}

__global__ void closs_finalize(const double* __restrict__ acc,
                               float* __restrict__ out,
                               double inv_count) {
    if (threadIdx.x == 0) out[0] = (float)(acc[0] * inv_count);
}

extern "C" void kernel_launch(void* const* d_in, const int* in_sizes, int n_in,
                              void* d_out, int out_size, void* d_ws, size_t ws_size,
                              hipStream_t stream) {
    const float* y_true = (const float*)d_in[0];
    const float* y_pred = (const float*)d_in[1];
    float* out = (float*)d_out;
    double* acc = (double*)d_ws;

    const long long total = (long long)in_sizes[0];
    const int n_img = (int)(total / ((long long)IMG_H * IMG_W));

    hipLaunchKernelGGL(closs_init, dim3(1), dim3(1), 0, stream, acc);

    dim3 grid(IMG_W / TILE_X, IMG_H / TILE_Y, n_img);
    hipLaunchKernelGGL(closs_main, grid, dim3(NTHREADS), 0, stream,
                       y_true, y_pred, acc);

    hipLaunchKernelGGL(closs_finalize, dim3(1), dim3(1), 0, stream,
                       acc, out, 1.0 / (double)total);
}